// DynamicsAttention_3934190043588
// MI455X (gfx1250) — compile-verified
//
#include <hip/hip_runtime.h>
#include <cmath>

// Sizes fixed by setup_inputs(): B=2, T=8, C=64, W=H=32, NOBJ=4, DIM=64
#define BSTO 64      // bs * n_obj * T
#define BST  16      // bs * T (distinct xin batches)
#define CCH  64      // channels
#define CH2  32      // q/k channels
#define DIMV 64      // v channels
#define NPIX 1024    // w*h
#define LOG2E 1.4426950408889634f

typedef float v2f __attribute__((ext_vector_type(2)));
typedef float v8f __attribute__((ext_vector_type(8)));

__device__ __forceinline__ v8f wmma4(v2f a, v2f b, v8f c) {
  // D = A(16x4,f32) * B(4x16,f32) + C(16x16,f32)
  return __builtin_amdgcn_wmma_f32_16x16x4_f32(false, a, false, b, (short)0, c,
                                               false, false);
}

// raw v_exp_f32 / v_log_f32 (log2 domain), no denorm-range fixup
__device__ __forceinline__ float exp2r(float x) {
  return __builtin_amdgcn_exp2f(x);
}
__device__ __forceinline__ float log2r(float x) {
  return __builtin_amdgcn_logf(x);
}
// single v_max_num_f32 without clang's operand canonicalization (WMMA outputs
// are already canonical); non-volatile so it stays CSE/schedule-friendly
__device__ __forceinline__ float maxr(float a, float b) {
  float d;
  asm("v_max_num_f32 %0, %1, %2" : "=v"(d) : "v"(a), "v"(b));
  return d;
}

// ------- projection: Q = (wq @ xdyn + bq) * log2(e), stored as paired rows --
// Qp[b][i][n] = (q[2i][n], q[2i+1][n])  -> one b64 load per WMMA A fragment
// Pre-scaling by log2(e) puts S = Q.K directly in the base-2 exp domain.
__global__ __launch_bounds__(256) void proj_q_kernel(
    const float* __restrict__ xdyn, const float* __restrict__ wq,
    const float* __restrict__ bq, v2f* __restrict__ Qp) {
  __shared__ float wql[CH2 * CCH];
  __shared__ float bql[CH2];
  const int t = threadIdx.x;
  for (int i = t; i < CH2 * CCH; i += 256) wql[i] = wq[i] * LOG2E;
  if (t < CH2) bql[t] = bq[t] * LOG2E;
  __syncthreads();

  const int b = blockIdx.y;                 // 0..63
  const int n = blockIdx.x * 256 + t;       // 0..1023
  float x[CCH];
  const float* xb = xdyn + (size_t)b * CCH * NPIX + n;
#pragma unroll
  for (int c = 0; c < CCH; ++c) x[c] = xb[(size_t)c * NPIX];

  v2f* Qb = Qp + ((size_t)b * 16) * NPIX + n;
  for (int i = 0; i < 16; ++i) {
    float a0 = bql[2 * i], a1 = bql[2 * i + 1];
    const float* w0 = &wql[(2 * i) * CCH];
    const float* w1 = &wql[(2 * i + 1) * CCH];
#pragma unroll
    for (int c = 0; c < CCH; ++c) {
      a0 = fmaf(w0[c], x[c], a0);
      a1 = fmaf(w1[c], x[c], a1);
    }
    v2f q2; q2.x = a0; q2.y = a1;
    Qb[(size_t)i * NPIX] = q2;              // coalesced b64
  }
}

// ------- projection: K (paired rows) and V^T (paired pixel rows) ------------
// Kp[b][i][n] = (k[2i][n], k[2i+1][n])
// Vp[b][np][d] = (v[d][2np], v[d][2np+1])   (np = pixel pair)
__global__ __launch_bounds__(256) void proj_kv_kernel(
    const float* __restrict__ xin, const float* __restrict__ wk,
    const float* __restrict__ bk, const float* __restrict__ wv,
    const float* __restrict__ bv, v2f* __restrict__ Kp,
    v2f* __restrict__ Vp) {
  __shared__ float wkl[CH2 * CCH];
  __shared__ float wvl[DIMV * CCH];
  __shared__ float bkl[CH2];
  __shared__ float bvl[DIMV];
  const int t = threadIdx.x;
  for (int i = t; i < CH2 * CCH; i += 256) wkl[i] = wk[i];
  for (int i = t; i < DIMV * CCH; i += 256) wvl[i] = wv[i];
  if (t < CH2) bkl[t] = bk[t];
  if (t < DIMV) bvl[t] = bv[t];
  __syncthreads();

  const int b = blockIdx.y;                 // 0..15 (bs*T)
  const int n = blockIdx.x * 256 + t;
  float x[CCH];
  const float* xb = xin + (size_t)b * CCH * NPIX + n;
#pragma unroll
  for (int c = 0; c < CCH; ++c) x[c] = xb[(size_t)c * NPIX];

  v2f* Kb = Kp + ((size_t)b * 16) * NPIX + n;
  for (int i = 0; i < 16; ++i) {
    float a0 = bkl[2 * i], a1 = bkl[2 * i + 1];
    const float* w0 = &wkl[(2 * i) * CCH];
    const float* w1 = &wkl[(2 * i + 1) * CCH];
#pragma unroll
    for (int c = 0; c < CCH; ++c) {
      a0 = fmaf(w0[c], x[c], a0);
      a1 = fmaf(w1[c], x[c], a1);
    }
    v2f k2; k2.x = a0; k2.y = a1;
    Kb[(size_t)i * NPIX] = k2;
  }

  // V: this thread owns pixel n -> component (n&1) of pair (n>>1)
  float* Vf = (float*)Vp + ((((size_t)b * (NPIX / 2) + (n >> 1)) * DIMV) << 1)
              + (n & 1);
  for (int d = 0; d < DIMV; ++d) {
    float acc = bvl[d];
    const float* wr = &wvl[d * CCH];
#pragma unroll
    for (int c = 0; c < CCH; ++c) acc = fmaf(wr[c], x[c], acc);
    Vf[2 * d] = acc;
  }
}

// ------------------------------- fused attention -----------------------------
// grid: (8 m-tiles, 64 batches), 256 threads = 8 waves, wave owns 16 query rows
__global__ __launch_bounds__(256) void attn_kernel(
    const v2f* __restrict__ Qp, const v2f* __restrict__ Kp,
    const v2f* __restrict__ Vp, float* __restrict__ outp,
    float* __restrict__ attn) {
  // stride 18 -> b64-aligned A reads, 16-lane bank pattern conflict-free
  __shared__ __align__(16) float pstage[8][16 * 18];

  const int tid  = threadIdx.x;
  const int wave = tid >> 5;
  const int lane = tid & 31;
  const int lh   = lane >> 4;      // which 16-lane half
  const int l16  = lane & 15;
  const int b    = blockIdx.y;                       // 0..63
  const int b16  = ((b >> 5) << 3) | (b & 7);        // distinct xin batch
  const int mbase = blockIdx.x * 128 + wave * 16;

  const v2f* Qb = Qp + (size_t)b   * 16 * NPIX;
  const v2f* Kb = Kp + (size_t)b16 * 16 * NPIX;
  const v2f* Vb = Vp + (size_t)b16 * (NPIX / 2) * DIMV;

  // Q tile in A layout: chunk c, half lh -> paired rows (4c+2lh, 4c+2lh+1)
  const int mA = mbase + l16;
  v2f qa[8];
#pragma unroll
  for (int c = 0; c < 8; ++c)
    qa[c] = Qb[(size_t)(2 * c + lh) * NPIX + mA];

  // per-lane online softmax stats (no cross-lane traffic inside the loop)
  float lmax[8], lsum[8];
#pragma unroll
  for (int r = 0; r < 8; ++r) { lmax[r] = -INFINITY; lsum[r] = 0.f; }

  // ---- pass 1: per-lane online max/sum over all 64 key tiles ----
  // S is already in the base-2 domain (Q pre-scaled by log2 e).
  for (int j = 0; j < 64; ++j) {
    const int n = j * 16 + l16;
    v8f s = {0.f, 0.f, 0.f, 0.f, 0.f, 0.f, 0.f, 0.f};
#pragma unroll
    for (int c = 0; c < 8; ++c)
      s = wmma4(qa[c], Kb[(size_t)(2 * c + lh) * NPIX + n], s);
#pragma unroll
    for (int r = 0; r < 8; ++r) {
      const float t = s[r];
      const float nm = maxr(lmax[r], t);
      lsum[r] = fmaf(lsum[r], exp2r(lmax[r] - nm), exp2r(t - nm));
      lmax[r] = nm;
    }
  }

  // ---- single cross-lane merge of (max,sum) within each 16-lane half ----
  // rbias = -(rowmax + log2(rowsum)): p = exp2(s + rbias) is fully normalized.
  float rbias[8];
#pragma unroll
  for (int r = 0; r < 8; ++r) {
    float m = lmax[r], l = lsum[r];
#pragma unroll
    for (int mask = 1; mask <= 8; mask <<= 1) {
      const float om = __shfl_xor(m, mask, 32);
      const float ol = __shfl_xor(l, mask, 32);
      const float nm = maxr(m, om);
      l = fmaf(l, exp2r(m - nm), ol * exp2r(om - nm));
      m = nm;
    }
    rbias[r] = -(m + log2r(l));
  }

  // ---- pass 2: recompute S, write P (NT), accumulate O = P * V^T ----
  v8f oacc[4];
#pragma unroll
  for (int dt = 0; dt < 4; ++dt)
#pragma unroll
    for (int r = 0; r < 8; ++r) oacc[dt][r] = 0.f;

  float* attb = attn + (size_t)b * NPIX * NPIX;
  for (int j = 0; j < 64; ++j) {
    const int n = j * 16 + l16;
    v8f s = {0.f, 0.f, 0.f, 0.f, 0.f, 0.f, 0.f, 0.f};
#pragma unroll
    for (int c = 0; c < 8; ++c)
      s = wmma4(qa[c], Kb[(size_t)(2 * c + lh) * NPIX + n], s);
#pragma unroll
    for (int r = 0; r < 8; ++r) {
      const float p = exp2r(s[r] + rbias[r]);           // add + v_exp only
      const int mrow = mbase + r + lh * 8;              // C-layout row
      __builtin_nontemporal_store(p, &attb[(size_t)mrow * NPIX + j * 16 + l16]);
      pstage[wave][(r + lh * 8) * 18 + l16] = p;        // stage for transpose
    }
    __syncthreads();   // uniform trip count across all 8 waves
#pragma unroll
    for (int cc = 0; cc < 4; ++cc) {
      const int kk = 4 * cc + lh * 2;                   // A-layout K index
      const v2f a = *(const v2f*)&pstage[wave][l16 * 18 + kk];
      const int np = j * 8 + 2 * cc + lh;               // paired pixel rows
#pragma unroll
      for (int dt = 0; dt < 4; ++dt) {
        const v2f vb = Vb[(size_t)np * DIMV + dt * 16 + l16];
        oacc[dt] = wmma4(a, vb, oacc[dt]);
      }
    }
    __syncthreads();
  }

  // epilogue: out[b][d][m] = O[m][d]
  float* ob = outp + (size_t)b * DIMV * NPIX;
#pragma unroll
  for (int dt = 0; dt < 4; ++dt)
#pragma unroll
    for (int r = 0; r < 8; ++r) {
      const int mrow = mbase + r + lh * 8;
      const int d = dt * 16 + l16;
      __builtin_nontemporal_store(oacc[dt][r], &ob[(size_t)d * NPIX + mrow]);
    }
}

extern "C" void kernel_launch(void* const* d_in, const int* in_sizes, int n_in,
                              void* d_out, int out_size, void* d_ws,
                              size_t ws_size, hipStream_t stream) {
  const float* inputs = (const float*)d_in[0];   // [2,8,64,32,32]
  const float* dyn    = (const float*)d_in[1];   // [2,4,8,64,32,32]
  const float* wq     = (const float*)d_in[2];
  const float* bq     = (const float*)d_in[3];
  const float* wk     = (const float*)d_in[4];
  const float* bk     = (const float*)d_in[5];
  const float* wv     = (const float*)d_in[6];
  const float* bv     = (const float*)d_in[7];

  float* out  = (float*)d_out;                       // [64,64,32,32]
  float* attn = out + (size_t)BSTO * DIMV * NPIX;    // [64,1024,1024]

  // workspace (paired-row layouts): Qp(8MB) + Kp(2MB) + Vp(4MB) ~= 14 MB
  v2f* Qp = (v2f*)d_ws;                              // [64][16][1024]
  v2f* Kp = Qp + (size_t)BSTO * 16 * NPIX;           // [16][16][1024]
  v2f* Vp = Kp + (size_t)BST * 16 * NPIX;            // [16][512][64]

  proj_q_kernel <<<dim3(4, BSTO), 256, 0, stream>>>(dyn, wq, bq, Qp);
  proj_kv_kernel<<<dim3(4, BST),  256, 0, stream>>>(inputs, wk, bk, wv, bv,
                                                    Kp, Vp);
  attn_kernel   <<<dim3(8, BSTO), 256, 0, stream>>>(Qp, Kp, Vp, out, attn);
}